// Distance_44607530336308
// MI455X (gfx1250) — compile-verified
//
#include <hip/hip_runtime.h>

// Problem constants (from reference): B=2, H=512, W=512, 6 output channels.
#define NB 2
#define NH 512
#define NW 512

// 2*sigma^2 for sigmas (6.4, 25.6, 51.2) -> reciprocals used as multipliers.
#define INV0 (1.0f / 81.92f)
#define INV1 (1.0f / 1310.72f)
#define INV2 (1.0f / 5242.88f)

// Vector type matching the async-to-LDS b128 builtin's pointee (V4i).
typedef int v4i __attribute__((ext_vector_type(4)));
typedef __attribute__((address_space(1))) v4i* g_v4i_ptr;  // global
typedef __attribute__((address_space(3))) v4i* l_v4i_ptr;  // LDS

// ---------------------------------------------------------------------------
// Pass 1: per-column 1-D EDT for both source values (tri==0 and tri==255).
// One thread per column; 256 adjacent columns per block so every trimap load
// and every g^2 store is fully coalesced. Writes g^2 (float) to workspace:
//   g2[((v*NB + b)*NH + h)*NW + w]
// ---------------------------------------------------------------------------
__global__ __launch_bounds__(256) void edt_cols_kernel(const int* __restrict__ tri,
                                                       float* __restrict__ g2) {
  const int b = blockIdx.x >> 1;                     // NW/256 == 2 chunks per batch
  const int w = ((blockIdx.x & 1) << 8) + threadIdx.x;
  const int base = b * NH * NW + w;
  float* __restrict__ g0 = g2 + (0 * NB + b) * NH * NW + w;
  float* __restrict__ g1 = g2 + (1 * NB + b) * NH * NW + w;

  // Downward scan: nearest source at-or-above (cummax of mask?h:-H).
  int f0 = -NH, f1 = -NH;
  for (int h = 0; h < NH; ++h) {
    const int t = tri[base + h * NW];
    if (t == 0)   f0 = h;
    if (t == 255) f1 = h;
    g0[h * NW] = (float)(h - f0);   // temporarily hold up-distance
    g1[h * NW] = (float)(h - f1);
  }
  // Upward scan: nearest source at-or-below; combine, square.
  int d0 = 2 * NH, d1 = 2 * NH;
  for (int h = NH - 1; h >= 0; --h) {
    const int t = tri[base + h * NW];
    if (t == 0)   d0 = h;
    if (t == 255) d1 = h;
    const float a0 = fminf(g0[h * NW], (float)(d0 - h));
    const float a1 = fminf(g1[h * NW], (float)(d1 - h));
    g0[h * NW] = a0 * a0;           // exact in f32 (max ~1.05e6 < 2^24)
    g1[h * NW] = a1 * a1;
  }
}

// Emit the 3 Gaussian channels for one (pixel, value): round half-to-even
// (rintf == jnp.round), clip to [0,255], store as float (uint8 semantics).
__device__ __forceinline__ void emit3(float* __restrict__ o, float d2) {
  o[0] = fminf(rintf(expf(-d2 * INV0) * 255.0f), 255.0f);
  o[1] = fminf(rintf(expf(-d2 * INV1) * 255.0f), 255.0f);
  o[2] = fminf(rintf(expf(-d2 * INV2) * 255.0f), 255.0f);
}

// ---------------------------------------------------------------------------
// Pass 2: exact min-plus over parabolas along W, out of LDS.
// One block per (b,h) row; stages both g^2 rows (value 0 / value 255, 2 KB
// each) into LDS with CDNA5 async global->LDS b128 transfers (ASYNCcnt),
// then each thread computes 2 pixels x 2 values over the 512-wide window.
// ---------------------------------------------------------------------------
__global__ __launch_bounds__(256) void edt_rows_kernel(const float* __restrict__ g2,
                                                       float* __restrict__ out) {
  __shared__ float S[2][NW];

  const int bh = blockIdx.x;               // b*NH + h
  const int tid = threadIdx.x;
  const int rowbase = bh * NW;             // v=0 row; v=1 row is +NB*NH*NW

  // Stage: 256 threads x one 16B chunk = 2 rows x 512 floats.
  {
    const int v = tid >> 7;                // 0..1
    const int c = (tid & 127) << 2;        // float offset within row
    const float* gsrc = g2 + rowbase + v * (NB * NH * NW) + c;
    float* ldst = &S[v][c];
#if __has_builtin(__builtin_amdgcn_global_load_async_to_lds_b128)
    __builtin_amdgcn_global_load_async_to_lds_b128(
        (g_v4i_ptr)(v4i*)(void*)const_cast<float*>(gsrc),
        (l_v4i_ptr)(v4i*)(void*)ldst,
        /*offset=*/0, /*cpol=*/0);
#if __has_builtin(__builtin_amdgcn_s_wait_asynccnt)
    __builtin_amdgcn_s_wait_asynccnt(0);
#else
    asm volatile("s_wait_asynccnt 0" ::: "memory");
#endif
#else
    *(float4*)ldst = *(const float4*)gsrc;
#endif
  }
  __syncthreads();

  // Min-plus: thread owns pixels wA=tid and wB=tid+256, both values.
  float dA0 = 3.4e38f, dA1 = 3.4e38f, dB0 = 3.4e38f, dB1 = 3.4e38f;
  float da = (float)tid;                   // wA - k, decremented per iter
  float db = (float)(tid + 256);           // wB - k
#pragma unroll 8
  for (int k = 0; k < NW; ++k) {
    const float s0 = S[0][k];
    const float s1 = S[1][k];
    const float qa = da * da;
    const float qb = db * db;
    dA0 = fminf(dA0, qa + s0);
    dA1 = fminf(dA1, qa + s1);
    dB0 = fminf(dB0, qb + s0);
    dB1 = fminf(dB1, qb + s1);
    da -= 1.0f;
    db -= 1.0f;
  }

  // Channels: value 0 -> c=0..2, value 255 -> c=3..5.
  float* oA = out + ((size_t)rowbase + tid) * 6;
  float* oB = out + ((size_t)rowbase + tid + 256) * 6;
  emit3(oA + 0, dA0);
  emit3(oA + 3, dA1);
  emit3(oB + 0, dB0);
  emit3(oB + 3, dB1);
}

extern "C" void kernel_launch(void* const* d_in, const int* in_sizes, int n_in,
                              void* d_out, int out_size, void* d_ws, size_t ws_size,
                              hipStream_t stream) {
  (void)in_sizes; (void)n_in; (void)out_size; (void)ws_size;
  const int* tri = (const int*)d_in[0];     // [B,H,W,1] int32 trimap
  float* out = (float*)d_out;               // [B,H,W,6] float32
  float* g2 = (float*)d_ws;                 // 2*NB*NH*NW floats = 4 MB scratch

  edt_cols_kernel<<<NB * (NW / 256), 256, 0, stream>>>(tri, g2);
  edt_rows_kernel<<<NB * NH, 256, 0, stream>>>(g2, out);
}